// FCBoltzmannModel_26783416058177
// MI455X (gfx1250) — compile-verified
//
#include <hip/hip_runtime.h>
#include <hip/hip_bf16.h>

// ---------------------------------------------------------------------------
// Problem sizes (fixed by the reference)
// ---------------------------------------------------------------------------
#define BATCH   2048
#define N_IN    1024
#define N_H     1024
#define N_OUT   512
#define N_UPD   (N_H + N_OUT)             // 1536
#define MAX_STEPS 2
#define N_TOTAL_UPD (MAX_STEPS * N_UPD)   // 3072
#define OUT_COLS (N_IN + N_H + N_OUT)     // 2560

typedef __attribute__((ext_vector_type(2))) float v2f;
typedef __attribute__((ext_vector_type(8))) float v8f;

// ---------------------------------------------------------------------------
// Kernel 1: f32 WMMA GEMM.  C[M x N] = A[M x K] * B (+ bias[n])
//   BT=true : B stored as Bt[N x K] row-major (i.e. C = A * Bt^T)
//   BT=false: B stored as B [K x N] row-major (i.e. C = A * B)
// One wave computes one 16x16 tile via V_WMMA_F32_16X16X4_F32.
// ---------------------------------------------------------------------------
template <bool BT>
__global__ __launch_bounds__(32) void gemm_f32_wmma(
    const float* __restrict__ A, const float* __restrict__ B,
    const float* __restrict__ bias, float* __restrict__ C,
    int M, int N, int K)
{
    const int tn   = blockIdx.x;          // tile col
    const int tm   = blockIdx.y;          // tile row
    const int lane = threadIdx.x;         // 0..31 (wave32)
    const int half = lane >> 4;           // 0 or 1
    const int mr   = lane & 15;

    const int rowA = tm * 16 + mr;        // M index for A fragment
    const int colB = tn * 16 + mr;        // N index for B fragment

    v8f acc = {};
    for (int k0 = 0; k0 < K; k0 += 4) {
        const int ka = k0 + 2 * half;
        v2f a, b;
        a.x = A[(size_t)rowA * K + ka];
        a.y = A[(size_t)rowA * K + ka + 1];
        if (BT) {
            b.x = B[(size_t)colB * K + ka];
            b.y = B[(size_t)colB * K + ka + 1];
        } else {
            b.x = B[(size_t)ka * N + colB];
            b.y = B[(size_t)(ka + 1) * N + colB];
        }
        // 8 args: (neg_a, A, neg_b, B, c_mod, C, reuse_a, reuse_b)
        acc = __builtin_amdgcn_wmma_f32_16x16x4_f32(
            false, a, false, b, (short)0, acc, false, false);
    }

    const int col = tn * 16 + mr;
    const float bv = bias ? bias[col] : 0.0f;
#pragma unroll
    for (int v = 0; v < 8; ++v) {
        const int row = tm * 16 + v + 8 * half;
        C[(size_t)row * N + col] = acc[v] + bv;
    }
}

// ---------------------------------------------------------------------------
// Kernel 2: transpose W2 [N_OUT x N_H] -> W2T [N_H x N_OUT] so the hidden-
// unit rank-1 update reads a contiguous, lane-coalesced row.
// ---------------------------------------------------------------------------
__global__ __launch_bounds__(256) void transpose_w2(
    const float* __restrict__ W2, float* __restrict__ W2T)
{
    int idx = blockIdx.x * 256 + threadIdx.x;          // over N_OUT*N_H
    int m = idx >> 10;            // / N_H
    int n = idx & (N_H - 1);      // % N_H
    W2T[(size_t)n * N_OUT + m] = W2[idx];
}

// ---------------------------------------------------------------------------
// Kernel 3: sequential Gibbs chain.  One wave per batch row (independent
// Markov chain).  Per-row fields live in LDS:
//   lH[n] = (x@W1^T + b1 + s2@W2)[row,n]   (G1 folded in at load)
//   lO[m] = (s1@W2^T + b2)[row,m]
// Unit updates become rank-1 LDS updates.  Critical-path trims:
//   * ids staged in LDS (broadcast reads)
//   * u batched 32-at-a-time per lane, broadcast via wave32 __shfl
//   * next update's W2/W2T row prefetched (global_prefetch_b8)
// LDS: 8*(1536*4 + 48*4) + 3072*4 = 62976 B.
// ---------------------------------------------------------------------------
#define ROWS_PB 8

__global__ __launch_bounds__(256) void gibbs_chain(
    const float* __restrict__ x,
    const float* __restrict__ s1_init,
    const float* __restrict__ s2_init,
    const float* __restrict__ W2,     // [N_OUT x N_H] row-major
    const float* __restrict__ W2T,    // [N_H x N_OUT]
    const float* __restrict__ u,      // [N_TOTAL_UPD x BATCH]
    const int*   __restrict__ ids,    // [N_TOTAL_UPD]
    const float* __restrict__ G1,     // [BATCH x N_H]  x@W1^T + b1
    const float* __restrict__ H0,     // [BATCH x N_H]  s2_init@W2
    const float* __restrict__ O0,     // [BATCH x N_OUT] s1_init@W2^T + b2
    float* __restrict__ out)          // [BATCH x OUT_COLS]
{
    __shared__ float    lH[ROWS_PB][N_H];
    __shared__ float    lO[ROWS_PB][N_OUT];
    __shared__ unsigned lS1[ROWS_PB][N_H / 32];
    __shared__ unsigned lS2[ROWS_PB][N_OUT / 32];
    __shared__ int      lIds[N_TOTAL_UPD];

    const int wave = threadIdx.x >> 5;
    const int lane = threadIdx.x & 31;
    const int row  = blockIdx.x * ROWS_PB + wave;

    // ---- stage ids once per block -----------------------------------------
    for (int i = threadIdx.x; i < N_TOTAL_UPD; i += 256) lIds[i] = ids[i];

    // ---- load per-row fields into LDS (fold G1 into lH) -------------------
    for (int i = lane; i < N_H; i += 32)
        lH[wave][i] = H0[(size_t)row * N_H + i] + G1[(size_t)row * N_H + i];
    for (int i = lane; i < N_OUT; i += 32)
        lO[wave][i] = O0[(size_t)row * N_OUT + i];
#pragma unroll
    for (int w = 0; w < N_H / 32; ++w) {
        float sv = s1_init[(size_t)row * N_H + w * 32 + lane];
        unsigned m = (unsigned)__ballot(sv > 0.5f);
        if (lane == 0) lS1[wave][w] = m;
    }
#pragma unroll
    for (int w = 0; w < N_OUT / 32; ++w) {
        float sv = s2_init[(size_t)row * N_OUT + w * 32 + lane];
        unsigned m = (unsigned)__ballot(sv > 0.5f);
        if (lane == 0) lS2[wave][w] = m;
    }
    __syncthreads();   // lIds visible to all waves

    // ---- sequential unit updates (independent chain per row/wave) ---------
    const float invT0 = 1.0f / 2.0f;                    // sweep 0: T = 2.0
    const float invT1 = 1.0f / 1.6374615061559636f;     // sweep 1: T = 2/e^{0.2}
    float uu_vec = 0.0f;
    for (int k = 0; k < N_TOTAL_UPD; ++k) {
        const int kb = k & 31;
        if (kb == 0) {
            // lane l pre-loads u for chain step k+l (N_TOTAL_UPD % 32 == 0)
            uu_vec = u[(size_t)(k + lane) * BATCH + row];
        }
        const float uu   = __shfl(uu_vec, kb, 32);
        const int   gid  = lIds[k];
        const float invT = (k < N_UPD) ? invT0 : invT1;

        // warm WGP$/L2 with next update's weight row while this one runs
        if (k + 1 < N_TOTAL_UPD) {
            const int g2 = lIds[k + 1];
            const char* pf = (g2 < N_H)
                ? (const char*)(W2T + (size_t)g2 * N_OUT) + lane * 64
                : (const char*)(W2 + (size_t)(g2 - N_H) * N_H) + lane * 128;
            __builtin_prefetch(pf, 0, 1);
        }

        if (gid < N_H) {                                 // ---- hidden unit n
            const int n = gid;
            const float gap = lH[wave][n];               // G1 already folded in
            const float p   = 1.0f / (1.0f + __expf(-gap * invT));
            const float act = (p > uu) ? 1.0f : 0.0f;
            const unsigned word = lS1[wave][n >> 5];
            const float old  = (float)((word >> (n & 31)) & 1u);
            const float dlt  = act - old;
            if (dlt != 0.0f) {                           // wave-uniform branch
                if (lane == 0) lS1[wave][n >> 5] = word ^ (1u << (n & 31));
                const float* w2c = W2T + (size_t)n * N_OUT;  // coalesced
#pragma unroll
                for (int m = lane; m < N_OUT; m += 32)
                    lO[wave][m] += dlt * w2c[m];
            }
        } else {                                         // ---- output unit m2
            const int m2 = gid - N_H;
            const float gap = lO[wave][m2];              // includes b2
            const float p   = 1.0f / (1.0f + __expf(-gap * invT));
            const float act = (p > uu) ? 1.0f : 0.0f;
            const unsigned word = lS2[wave][m2 >> 5];
            const float old = (float)((word >> (m2 & 31)) & 1u);
            const float dlt = act - old;
            if (dlt != 0.0f) {
                if (lane == 0) lS2[wave][m2 >> 5] = word ^ (1u << (m2 & 31));
                const float* w2r = W2 + (size_t)m2 * N_H;    // coalesced
#pragma unroll
                for (int n = lane; n < N_H; n += 32)
                    lH[wave][n] += dlt * w2r[n];
            }
        }
    }

    // ---- emit concat([x, s1, s2]) -----------------------------------------
    float* orow = out + (size_t)row * OUT_COLS;
    for (int i = lane; i < N_IN; i += 32)
        orow[i] = x[(size_t)row * N_IN + i];
#pragma unroll
    for (int w = 0; w < N_H / 32; ++w) {
        unsigned word = lS1[wave][w];
        orow[N_IN + w * 32 + lane] = (float)((word >> lane) & 1u);
    }
#pragma unroll
    for (int w = 0; w < N_OUT / 32; ++w) {
        unsigned word = lS2[wave][w];
        orow[N_IN + N_H + w * 32 + lane] = (float)((word >> lane) & 1u);
    }
}

// ---------------------------------------------------------------------------
// Host-side launch.  Inputs (setup_inputs order):
//   0:x 1:s1_init 2:s2_init 3:W1 4:b1 5:W2 6:b2 7:u 8:ids
// Workspace: G1 (8MB) | H0 (8MB) | O0 (4MB) | W2T (2MB)
// ---------------------------------------------------------------------------
extern "C" void kernel_launch(void* const* d_in, const int* in_sizes, int n_in,
                              void* d_out, int out_size, void* d_ws, size_t ws_size,
                              hipStream_t stream)
{
    const float* x  = (const float*)d_in[0];
    const float* s1 = (const float*)d_in[1];
    const float* s2 = (const float*)d_in[2];
    const float* W1 = (const float*)d_in[3];
    const float* b1 = (const float*)d_in[4];
    const float* W2 = (const float*)d_in[5];
    const float* b2 = (const float*)d_in[6];
    const float* u  = (const float*)d_in[7];
    const int*  ids = (const int*)d_in[8];
    float* out = (float*)d_out;

    float* G1  = (float*)d_ws;
    float* H0  = G1 + (size_t)BATCH * N_H;
    float* O0  = H0 + (size_t)BATCH * N_H;
    float* W2T = O0 + (size_t)BATCH * N_OUT;

    dim3 wblk(32);
    // G1 = x @ W1^T + b1       (A=x, Bt=W1)
    gemm_f32_wmma<true ><<<dim3(N_H   / 16, BATCH / 16), wblk, 0, stream>>>(
        x,  W1, b1,      G1, BATCH, N_H,   N_IN);
    // H0 = s2_init @ W2        (A=s2, B=W2 non-transposed)
    gemm_f32_wmma<false><<<dim3(N_H   / 16, BATCH / 16), wblk, 0, stream>>>(
        s2, W2, nullptr, H0, BATCH, N_H,   N_OUT);
    // O0 = s1_init @ W2^T + b2 (A=s1, Bt=W2)
    gemm_f32_wmma<true ><<<dim3(N_OUT / 16, BATCH / 16), wblk, 0, stream>>>(
        s1, W2, b2,      O0, BATCH, N_OUT, N_H);

    transpose_w2<<<(N_OUT * N_H) / 256, 256, 0, stream>>>(W2, W2T);

    gibbs_chain<<<BATCH / ROWS_PB, 256, 0, stream>>>(
        x, s1, s2, W2, W2T, u, ids, G1, H0, O0, out);
}